// EmbeddingModel_13142599925845
// MI455X (gfx1250) — compile-verified
//
#include <hip/hip_runtime.h>
#include <hip/hip_bf16.h>

#define HD 128

typedef float v2f __attribute__((ext_vector_type(2)));
typedef float v8f __attribute__((ext_vector_type(8)));

__device__ __forceinline__ v8f wmma_f32(v2f a, v2f b, v8f c) {
#if __has_builtin(__builtin_amdgcn_wmma_f32_16x16x4_f32)
  // 8 args: (neg_a, A, neg_b, B, c_mod, C, reuse_a, reuse_b)
  return __builtin_amdgcn_wmma_f32_16x16x4_f32(false, a, false, b, (short)0, c,
                                               false, false);
#else
  c[0] += a.x * b.x + a.y * b.y;  // keeps TU compiling; histogram shows wmma==0
  return c;
#endif
}

#if __has_builtin(__builtin_amdgcn_tensor_load_to_lds) && \
    __has_builtin(__builtin_amdgcn_s_wait_tensorcnt)
#define SAGE_TDM 1
typedef int v4i __attribute__((ext_vector_type(4)));
typedef int v8i __attribute__((ext_vector_type(8)));
typedef unsigned int v4u __attribute__((ext_vector_type(4)));

// Issue one TDM 2D tile load: a rows x dim0 f32 tile from global memory (row-major, stride=dim0)
// into LDS at the address in lds_addr. Tracked by TENSORcnt; EXEC ignored. D# per ISA 08 §8.
// clang-23 toolchain form: (v4u g0, v8i g1, v4i g2, v4i g3, v8i, i32 cpol).
__device__ __forceinline__ void tdm_load_tile_f32(unsigned lds_addr,
                                                  const float* gptr, int dim0,
                                                  int rows_remaining,
                                                  int tile_rows) {
  unsigned long long ga = (unsigned long long)(size_t)gptr;
  v4u g0 = {1u,                                  // count=1 (valid), user mode
            lds_addr,                            // lds_addr [63:32]
            (unsigned)(ga & 0xFFFFFFFFull),      // global_addr [95:64]
            (unsigned)(((ga >> 32) & 0x1FFFFFFull)  // global_addr [120:96]
                       | (2u << 30))};           // type=2 ("image") [127:126]
  unsigned rows = (unsigned)(rows_remaining > 0 ? rows_remaining : 0);
  v8i g1 = {(int)(2u << 16),                          // data_size=2 (4B)
            (int)((unsigned)dim0 << 16),              // tensor_dim0[15:0]@[63:48]
            (int)((rows & 0xFFFFu) << 16),            // dim0 hi=0, tensor_dim1 lo
            (int)(((rows >> 16) & 0xFFFFu)            // tensor_dim1 hi
                  | ((unsigned)dim0 << 16)),          // tile_dim0 @[127:112]
            (int)(unsigned)tile_rows,                 // tile_dim1 @[143:128]
            dim0,                                     // tensor_dim0_stride lo32
            0, 0};
  v4i gz4 = {0, 0, 0, 0};                             // 2D: groups 2/3 zeroed
  v8i gz8 = {0, 0, 0, 0, 0, 0, 0, 0};
  __builtin_amdgcn_tensor_load_to_lds(g0, g1, gz4, gz4, gz8, 0);
}
#endif

// --------------------------------------------------------------------------
// xu = user_x @ W(5x128) + b + emb[ids]   (K=5: scalar, memory bound)
__global__ void k_input_user(const float* __restrict__ x,
                             const float* __restrict__ W,
                             const float* __restrict__ b,
                             const float* __restrict__ emb,
                             const int* __restrict__ ids,
                             float* __restrict__ out, int NU) {
  int gid = blockIdx.x * blockDim.x + threadIdx.x;
  if (gid >= NU * HD) return;
  int n = gid >> 7;
  int c = gid & (HD - 1);
  float v = b[c] + emb[(size_t)ids[n] * HD + c];
  const float* xr = x + (size_t)n * 5;
#pragma unroll
  for (int k = 0; k < 5; ++k) v += xr[k] * W[k * HD + c];
  out[gid] = v;
}

// --------------------------------------------------------------------------
// OUT[N,128] = A[N,K] @ W[K,128] + bias + emb[ids]   (K multiple of 128)
__global__ void k_gemm_bias_emb(const float* __restrict__ A, int K,
                                const float* __restrict__ W,
                                const float* __restrict__ bias,
                                const float* __restrict__ emb,
                                const int* __restrict__ ids,
                                float* __restrict__ out, int N) {
  __shared__ float sA[16][HD];
  const int tid = threadIdx.x;
  const int row0 = blockIdx.x * 16;
  const int wave = tid >> 5, lane = tid & 31;
  const int col0 = wave << 4;
  const int m = lane & 15;
  const int kh = (lane >> 4) << 1;
  v8f acc = {};
  for (int kc = 0; kc < K; kc += HD) {
    __syncthreads();
    for (int i = tid; i < 16 * (HD / 4); i += 256) {
      int r = i >> 5;
      int c4 = (i & 31) << 2;
      int rr = row0 + r; if (rr >= N) rr = N - 1;
      *(float4*)&sA[r][c4] = *(const float4*)&A[(size_t)rr * K + kc + c4];
    }
    if (kc + HD < K)  // speculative prefetch -> global_prefetch_b8
      __builtin_prefetch(&A[(size_t)(row0 + (tid & 15)) * K + kc + HD], 0, 0);
    __syncthreads();
#pragma unroll 8
    for (int k0 = 0; k0 < HD; k0 += 4) {
      v2f a, b;
      a.x = sA[m][k0 + kh];
      a.y = sA[m][k0 + kh + 1];
      b.x = W[(size_t)(kc + k0 + kh) * HD + col0 + m];
      b.y = W[(size_t)(kc + k0 + kh + 1) * HD + col0 + m];
      acc = wmma_f32(a, b, acc);
    }
  }
  const int half = lane >> 4;
  const int colg = col0 + m;
  const float bb = bias[colg];
#pragma unroll
  for (int g = 0; g < 8; ++g) {
    int r = row0 + g + half * 8;
    if (r < N)
      out[(size_t)r * HD + colg] =
          acc[g] + bb + emb[(size_t)ids[r] * HD + colg];
  }
}

// --------------------------------------------------------------------------
// OUT = (S ./ max(deg,1)) @ WL + bias + X @ WR   [optional relu]
// A-tiles staged to LDS by the Tensor Data Mover (TENSORcnt), degree
// normalization fused as an in-LDS scale pass.
__global__ void k_sage(const float* __restrict__ S, const float* __restrict__ deg,
                       const float* __restrict__ WL, const float* __restrict__ bias,
                       const float* __restrict__ X, const float* __restrict__ WR,
                       float* __restrict__ out, int N, int relu) {
  __shared__ float sM[16][HD];
  __shared__ float sX[16][HD];
  const int tid = threadIdx.x;
  const int row0 = blockIdx.x * 16;
#if defined(SAGE_TDM)
  if (tid == 0) {  // one wave issues both DMAs; EXEC ignored by TDM
    tdm_load_tile_f32((unsigned)(size_t)&sM[0][0], S + (size_t)row0 * HD, HD,
                      N - row0, 16);
    tdm_load_tile_f32((unsigned)(size_t)&sX[0][0], X + (size_t)row0 * HD, HD,
                      N - row0, 16);
    __builtin_amdgcn_s_wait_tensorcnt(0);
  }
  __syncthreads();
  // fuse mean-normalization: sM[r][:] *= 1/max(deg,1) in LDS
  for (int i = tid; i < 16 * (HD / 4); i += 256) {
    int r = i >> 5;
    int c4 = (i & 31) << 2;
    int rr = row0 + r; if (rr >= N) rr = N - 1;
    float inv = 1.0f / fmaxf(deg[rr], 1.0f);
    float4 v = *(float4*)&sM[r][c4];
    v.x *= inv; v.y *= inv; v.z *= inv; v.w *= inv;
    *(float4*)&sM[r][c4] = v;
  }
#else
  for (int i = tid; i < 16 * (HD / 4); i += 256) {
    int r = i >> 5;
    int c4 = (i & 31) << 2;
    int rr = row0 + r; if (rr >= N) rr = N - 1;
    float inv = 1.0f / fmaxf(deg[rr], 1.0f);
    float4 s4 = *(const float4*)&S[(size_t)rr * HD + c4];
    s4.x *= inv; s4.y *= inv; s4.z *= inv; s4.w *= inv;
    *(float4*)&sM[r][c4] = s4;
    *(float4*)&sX[r][c4] = *(const float4*)&X[(size_t)rr * HD + c4];
  }
#endif
  __syncthreads();
  const int wave = tid >> 5, lane = tid & 31;
  const int col0 = wave << 4;
  const int m = lane & 15;
  const int kh = (lane >> 4) << 1;
  v8f acc = {};
#pragma unroll 4
  for (int k0 = 0; k0 < HD; k0 += 4) {
    v2f a1, b1, a2, b2;
    a1.x = sM[m][k0 + kh];
    a1.y = sM[m][k0 + kh + 1];
    b1.x = WL[(size_t)(k0 + kh) * HD + col0 + m];
    b1.y = WL[(size_t)(k0 + kh + 1) * HD + col0 + m];
    acc = wmma_f32(a1, b1, acc);
    a2.x = sX[m][k0 + kh];
    a2.y = sX[m][k0 + kh + 1];
    b2.x = WR[(size_t)(k0 + kh) * HD + col0 + m];
    b2.y = WR[(size_t)(k0 + kh + 1) * HD + col0 + m];
    acc = wmma_f32(a2, b2, acc);
  }
  const int half = lane >> 4;
  const int colg = col0 + m;
  const float bb = bias[colg];
#pragma unroll
  for (int g = 0; g < 8; ++g) {
    int r = row0 + g + half * 8;
    if (r < N) {
      float v = acc[g] + bb;
      if (relu) v = fmaxf(v, 0.0f);
      out[(size_t)r * HD + colg] = v;
    }
  }
}

// --------------------------------------------------------------------------
__global__ void k_degree(const int* __restrict__ src, const int* __restrict__ dst,
                         float* __restrict__ deg_u, float* __restrict__ deg_r,
                         int E) {
  int e = blockIdx.x * blockDim.x + threadIdx.x;
  if (e >= E) return;
  atomicAdd(&deg_u[src[e]], 1.0f);
  atomicAdd(&deg_r[dst[e]], 1.0f);
}

// --------------------------------------------------------------------------
// scatter-sum: acc[didx[e]] += X[sidx[e]]  (32 lanes/edge, float4 each;
// accumulators are L2-resident, fp32 global atomics)
__global__ void k_scatter(const float* __restrict__ X,
                          const int* __restrict__ sidx,
                          const int* __restrict__ didx,
                          float* __restrict__ acc, int E) {
  long long gid = (long long)blockIdx.x * blockDim.x + threadIdx.x;
  if (gid >= (long long)E * 32) return;
  int e = (int)(gid >> 5);
  int c4 = ((int)gid & 31) << 2;
  int s = sidx[e], d = didx[e];
  const float4 v = *(const float4*)&X[(size_t)s * HD + c4];
  float* p = &acc[(size_t)d * HD + c4];
  atomicAdd(p + 0, v.x);
  atomicAdd(p + 1, v.y);
  atomicAdd(p + 2, v.z);
  atomicAdd(p + 3, v.w);
}

// --------------------------------------------------------------------------
__global__ void k_dot(const float* __restrict__ XU, const float* __restrict__ XR,
                      const int* __restrict__ ls, const int* __restrict__ ld,
                      float* __restrict__ out, int L) {
  long long gid = (long long)blockIdx.x * blockDim.x + threadIdx.x;
  if (gid >= (long long)L * 32) return;
  int l = (int)(gid >> 5);
  int lane = threadIdx.x & 31;
  int c4 = lane << 2;
  const float4 a = *(const float4*)&XU[(size_t)ls[l] * HD + c4];
  const float4 b = *(const float4*)&XR[(size_t)ld[l] * HD + c4];
  float s = a.x * b.x + a.y * b.y + a.z * b.z + a.w * b.w;
#pragma unroll
  for (int off = 16; off > 0; off >>= 1) s += __shfl_down(s, off, 32);
  if (lane == 0) out[l] = s;
}

// ==========================================================================
extern "C" void kernel_launch(void* const* d_in, const int* in_sizes, int n_in,
                              void* d_out, int out_size, void* d_ws,
                              size_t ws_size, hipStream_t stream) {
  const float* user_lin_w = (const float*)d_in[0];
  const float* user_lin_b = (const float*)d_in[1];
  const float* rest_lin_w = (const float*)d_in[2];
  const float* rest_lin_b = (const float*)d_in[3];
  const float* user_emb   = (const float*)d_in[4];
  const float* rest_emb   = (const float*)d_in[5];
  const float* c1_rat_wl = (const float*)d_in[6];
  const float* c1_rat_wr = (const float*)d_in[7];
  const float* c1_rat_b  = (const float*)d_in[8];
  const float* c1_rev_wl = (const float*)d_in[9];
  const float* c1_rev_wr = (const float*)d_in[10];
  const float* c1_rev_b  = (const float*)d_in[11];
  const float* c2_rat_wl = (const float*)d_in[12];
  const float* c2_rat_wr = (const float*)d_in[13];
  const float* c2_rat_b  = (const float*)d_in[14];
  const float* c2_rev_wl = (const float*)d_in[15];
  const float* c2_rev_wr = (const float*)d_in[16];
  const float* c2_rev_b  = (const float*)d_in[17];
  const float* user_x = (const float*)d_in[18];
  const float* rest_x = (const float*)d_in[19];
  const int* user_node_id = (const int*)d_in[20];
  const int* rest_node_id = (const int*)d_in[21];
  const int* rating_src = (const int*)d_in[22];
  const int* rating_dst = (const int*)d_in[23];
  const int* label_src  = (const int*)d_in[24];
  const int* label_dst  = (const int*)d_in[25];

  const int NU = in_sizes[20];
  const int NR = in_sizes[21];
  const int E  = in_sizes[22];
  const int L  = in_sizes[24];
  const int Kr = in_sizes[19] / NR;   // 640

  float* ws = (float*)d_ws;
  float* xu_a = ws; ws += (size_t)NU * HD;
  float* xu_b = ws; ws += (size_t)NU * HD;
  float* xr_a = ws; ws += (size_t)NR * HD;
  float* xr_b = ws; ws += (size_t)NR * HD;
  float* acc  = ws; ws += (size_t)NU * HD;   // shared accumulator (NU >= NR)
  float* deg_u = ws; ws += NU;
  float* deg_r = ws; ws += NR;
  (void)ws_size; (void)n_in; (void)out_size;

  (void)hipMemsetAsync(deg_u, 0, (size_t)(NU + NR) * sizeof(float), stream);
  k_degree<<<(E + 255) / 256, 256, 0, stream>>>(rating_src, rating_dst,
                                                deg_u, deg_r, E);

  k_input_user<<<((size_t)NU * HD + 255) / 256, 256, 0, stream>>>(
      user_x, user_lin_w, user_lin_b, user_emb, user_node_id, xu_a, NU);
  k_gemm_bias_emb<<<(NR + 15) / 16, 256, 0, stream>>>(
      rest_x, Kr, rest_lin_w, rest_lin_b, rest_emb, rest_node_id, xr_a, NR);

  const long long totS = (long long)E * 32;
  const int nbS = (int)((totS + 255) / 256);

  auto layer = [&](const float* wl_rat, const float* wr_rat, const float* b_rat,
                   const float* wl_rev, const float* wr_rev, const float* b_rev,
                   const float* xu_in, const float* xr_in,
                   float* xu_out, float* xr_out, int relu) {
    (void)hipMemsetAsync(acc, 0, (size_t)NR * HD * sizeof(float), stream);
    k_scatter<<<nbS, 256, 0, stream>>>(xu_in, rating_src, rating_dst, acc, E);
    k_sage<<<(NR + 15) / 16, 256, 0, stream>>>(acc, deg_r, wl_rat, b_rat,
                                               xr_in, wr_rat, xr_out, NR, relu);
    (void)hipMemsetAsync(acc, 0, (size_t)NU * HD * sizeof(float), stream);
    k_scatter<<<nbS, 256, 0, stream>>>(xr_in, rating_dst, rating_src, acc, E);
    k_sage<<<(NU + 15) / 16, 256, 0, stream>>>(acc, deg_u, wl_rev, b_rev,
                                               xu_in, wr_rev, xu_out, NU, relu);
  };

  layer(c1_rat_wl, c1_rat_wr, c1_rat_b, c1_rev_wl, c1_rev_wr, c1_rev_b,
        xu_a, xr_a, xu_b, xr_b, 1);
  layer(c2_rat_wl, c2_rat_wr, c2_rat_b, c2_rev_wl, c2_rev_wr, c2_rev_b,
        xu_b, xr_b, xu_a, xr_a, 1);
  layer(c2_rat_wl, c2_rat_wr, c2_rat_b, c2_rev_wl, c2_rev_wr, c2_rev_b,
        xu_a, xr_a, xu_b, xr_b, 0);

  const long long totD = (long long)L * 32;
  k_dot<<<(int)((totD + 255) / 256), 256, 0, stream>>>(
      xu_b, xr_b, label_src, label_dst, (float*)d_out, L);
}